// BuildPatches_65309272703444
// MI455X (gfx1250) — compile-verified
//
#include <hip/hip_runtime.h>
#include <cmath>

typedef __attribute__((ext_vector_type(2))) float v2f;
typedef __attribute__((ext_vector_type(8))) float v8f;

#define BB   2
#define NN   16384
#define RR   4096
#define KK   32
#define EPSV 1e-5f

// ---------------------------------------------------------------------------
// Kernel 1: squared pairwise distance matrix via V_WMMA_F32_16X16X4_F32.
// One wave computes 16 (roots) x 256 (points) = sixteen 16x16 WMMA tiles.
// A row (16x4)  = (-2rx, -2ry, -2rz, |r|^2)   [lanes 0-15: K0,K1 ; 16-31: K2,K3]
// B col (4x16)  = ( px ,  py ,  pz ,   1  )   [same striping]
// C[m][n]       = |p_n|^2  (same value in all 8 accum VGPRs of a lane)
// => D = |r|^2 - 2 r.p + |p|^2  exactly, fp32.
// ---------------------------------------------------------------------------
__global__ __launch_bounds__(256) void sqdist_wmma_kernel(
    const float* __restrict__ points,   // [B,N,3]
    const float* __restrict__ roots,    // [B,R,3]
    float* __restrict__ sq)             // [B,R,N]
{
    const int wave = blockIdx.x * (blockDim.x >> 5) + (threadIdx.x >> 5);
    const int lane = threadIdx.x & 31;

    const int ntc = wave & 63;          // 64 chunks of 256 points along N
    const int rt  = (wave >> 6) & 255;  // 256 root tiles of 16
    const int b   = wave >> 14;         // batch

    // ---- A operand (constant across the 16 inner tiles) ----
    const int m = rt * 16 + (lane & 15);
    const float* rp = roots + ((size_t)b * RR + m) * 3;
    const float rx = rp[0], ry = rp[1], rz = rp[2];
    const bool hi = (lane & 16) != 0;
    v2f a;
    a.x = hi ? (-2.0f * rz) : (-2.0f * rx);
    a.y = hi ? (rx * rx + ry * ry + rz * rz) : (-2.0f * ry);

    const int rowBase = rt * 16 + (hi ? 8 : 0);
    float* const outRow = sq + ((size_t)b * RR + rowBase) * NN;

    for (int t = 0; t < 16; ++t) {
        const int n = ntc * 256 + t * 16 + (lane & 15);
        const float* pp = points + ((size_t)b * NN + n) * 3;
        const float px = pp[0], py = pp[1], pz = pp[2];
        const float p2 = px * px + py * py + pz * pz;

        v2f bv;
        bv.x = hi ? pz : px;
        bv.y = hi ? 1.0f : py;

        v8f c;
#pragma unroll
        for (int j = 0; j < 8; ++j) c[j] = p2;

        v8f d = __builtin_amdgcn_wmma_f32_16x16x4_f32(
            /*neg_a=*/false, a, /*neg_b=*/false, bv,
            /*c_mod=*/(short)0, c, /*reuse_a=*/false, /*reuse_b=*/false);

        float* o = outRow + n;
#pragma unroll
        for (int j = 0; j < 8; ++j) o[(size_t)j * NN] = d[j];
    }
}

// ---------------------------------------------------------------------------
// Kernel 2: one wave per root. Lane l scans columns l, l+32, ... (coalesced),
// keeping a register-resident max-replace top-32. Then 32 shfl_xor merge
// rounds extract the global top-32 in ascending (dist, idx) order, matching
// jax.lax.top_k tie-breaking, and emit patches / index pairs / distances.
// ---------------------------------------------------------------------------
__global__ __launch_bounds__(256) void topk_emit_kernel(
    const float* __restrict__ points,   // [B,N,3]
    const float* __restrict__ roots,    // [B,R,3]
    const float* __restrict__ sq,       // [B,R,N]
    float* __restrict__ patches,        // [B,R,K,3]
    float* __restrict__ idxout,         // [B,R,K,2] (as float)
    float* __restrict__ dist)           // [B,R,K]
{
    const int wave = blockIdx.x * (blockDim.x >> 5) + (threadIdx.x >> 5);
    const int lane = threadIdx.x & 31;
    const int r = wave & (RR - 1);
    const int b = wave >> 12;

    const float* row = sq + ((size_t)b * RR + r) * NN;
    const float* rt3 = roots + ((size_t)b * RR + r) * 3;
    const float rx = rt3[0], ry = rt3[1], rz = rt3[2];

    // ---- per-lane local top-32 over its strided 512-element subsequence ----
    float hd[KK];
    int   hx[KK];
#pragma unroll
    for (int j = 0; j < KK; ++j) {
        const int n = lane + 32 * j;
        hd[j] = row[n];
        hx[j] = n;
    }
    float worst = -3.0e38f; int wpos = 0;
#pragma unroll
    for (int j = 0; j < KK; ++j)
        if (hd[j] > worst) { worst = hd[j]; wpos = j; }

    for (int t = KK; t < NN / 32; ++t) {
        const int n = lane + 32 * t;
        const float d = row[n];
        if (d < worst) {
#pragma unroll
            for (int j = 0; j < KK; ++j)
                if (j == wpos) { hd[j] = d; hx[j] = n; }
            worst = -3.0e38f;
#pragma unroll
            for (int j = 0; j < KK; ++j)
                if (hd[j] > worst) { worst = hd[j]; wpos = j; }
        }
    }

    // ---- 32 rounds of wave-wide extract-min (lexicographic on (d, idx)) ----
    const size_t obase = (size_t)(b * RR + r) * KK;
    for (int k = 0; k < KK; ++k) {
        // local min of this lane's remaining candidates
        float mv = 3.0e38f; int mi = 0x7fffffff; int mpos = 0;
#pragma unroll
        for (int j = 0; j < KK; ++j) {
            const bool better = (hd[j] < mv) || (hd[j] == mv && hx[j] < mi);
            if (better) { mv = hd[j]; mi = hx[j]; mpos = j; }
        }

        // wave-wide lexicographic min reduction (all lanes converge)
        float gv = mv; int gi = mi;
#pragma unroll
        for (int off = 16; off > 0; off >>= 1) {
            const float ov = __shfl_xor(gv, off, 32);
            const int   oi = __shfl_xor(gi, off, 32);
            if (ov < gv || (ov == gv && oi < gi)) { gv = ov; gi = oi; }
        }

        // owning lane retires the winning slot
        if (mv == gv && mi == gi) {
#pragma unroll
            for (int j = 0; j < KK; ++j)
                if (j == mpos) { hd[j] = 3.0e38f; hx[j] = 0x7fffffff; }
        }

        // one lane emits outputs for this rank
        if (lane == (k & 31)) {
            dist[obase + k] = sqrtf(fmaxf(gv, EPSV));
            idxout[(obase + k) * 2 + 0] = (float)b;
            idxout[(obase + k) * 2 + 1] = (float)gi;
            const float* pp = points + ((size_t)b * NN + gi) * 3;
            float* po = patches + (obase + k) * 3;
            po[0] = pp[0] - rx;
            po[1] = pp[1] - ry;
            po[2] = pp[2] - rz;
        }
    }
}

extern "C" void kernel_launch(void* const* d_in, const int* in_sizes, int n_in,
                              void* d_out, int out_size, void* d_ws, size_t ws_size,
                              hipStream_t stream) {
    (void)in_sizes; (void)n_in; (void)out_size; (void)d_ws; (void)ws_size;

    const float* points = (const float*)d_in[0];  // [2,16384,3]
    const float* roots  = (const float*)d_in[1];  // [2,4096,3]
    float* out = (float*)d_out;

    // Output layout (flat float, reference return order):
    float* patches = out;                                        // B*R*K*3 = 786432
    float* idxout  = patches + (size_t)BB * RR * KK * 3;         // B*R*K*2 = 524288
    float* dist    = idxout  + (size_t)BB * RR * KK * 2;         // B*R*K   = 262144
    float* sqmat   = dist    + (size_t)BB * RR * KK;             // B*R*N   = 134217728

    // Kernel 1: 2 * 256 * 64 = 32768 waves, 8 waves (256 thr) per block.
    sqdist_wmma_kernel<<<32768 / 8, 256, 0, stream>>>(points, roots, sqmat);

    // Kernel 2: one wave per root -> 8192 waves, 8 per block.
    topk_emit_kernel<<<8192 / 8, 256, 0, stream>>>(points, roots, sqmat,
                                                   patches, idxout, dist);
}